// TCVAELoss_28870770163806
// MI455X (gfx1250) — compile-verified
//
#include <hip/hip_runtime.h>
#include <hip/hip_bf16.h>
#include <math.h>

typedef __attribute__((ext_vector_type(2))) float v2f;
typedef __attribute__((ext_vector_type(8))) float v8f;

constexpr int BB = 2048;
constexpr int DD = 64;
constexpr float kLog2Pi = 1.8378770664093453f;

// Workspace layout (units: floats)
constexpr size_t OFF_A   = 0;                           // [B][128]  = [x^2 | x]
constexpr size_t OFF_BW  = OFF_A  + (size_t)BB * 128;   // [B][128]  = [q | a]
constexpr size_t OFF_ET  = OFF_BW + (size_t)BB * 128;   // [D][B]    e transposed
constexpr size_t OFF_QT  = OFF_ET + (size_t)DD * BB;    // [D][B]
constexpr size_t OFF_AT  = OFF_QT + (size_t)DD * BB;    // [D][B]
constexpr size_t OFF_C   = OFF_AT + (size_t)DD * BB;    // [B]  c[j] = sum_d e[j,d]
constexpr size_t OFF_LQC = OFF_C   + BB;                // [B]  log q(z|x)
constexpr size_t OFF_LPZ = OFF_LQC + BB;                // [B]  log p(z)
constexpr size_t OFF_LQZ = OFF_LPZ + BB;                // [B]  log q(z)
constexpr size_t OFF_LPQ = OFF_LQZ + BB;                // [B]  log prod q(z_i)
constexpr size_t OFF_LW  = OFF_LPQ + BB;                // [16] {log(1/N), log(strat)}
constexpr size_t OFF_S   = OFF_LW  + 16;                // [B][B]  S + log_iw

__device__ __forceinline__ void lse_push(float& m, float& s, float v) {
    if (v <= m) {
        s += __expf(v - m);
    } else {
        s = s * __expf(m - v) + 1.0f;   // exp(-inf)=0 handles initial m=-inf
        m = v;
    }
}

__device__ __forceinline__ void lse_merge(float& m, float& s, float m2, float s2) {
    float M = fmaxf(m, m2);
    s = s * __expf(m - M) + s2 * __expf(m2 - M);
    m = M;
}

// ---------------------------------------------------------------------------
// Kernel 1: per-row preprocessing. One wave per row r; lanes cover 2 dims each.
// Builds GEMM operands, transposed per-dim tables, the 1-D reductions, and
// (block 0 / thread 0 only) the two log-importance-weight scalars, so the hot
// kernels never touch f64.
// ---------------------------------------------------------------------------
__global__ void k_prep(const float* __restrict__ x, const float* __restrict__ mu,
                       const float* __restrict__ lv, const int* __restrict__ dsize,
                       float* __restrict__ ws) {
    if (blockIdx.x == 0 && threadIdx.x == 0) {
        const double Nd = (double)(*dsize);
        ws[OFF_LW + 0] = (float)(-log(Nd));                                      // log(1/N)
        ws[OFF_LW + 1] = (float)log((Nd - (double)BB) / (Nd * (double)(BB - 1))); // log strat
    }

    const int lane = threadIdx.x & 31;
    const int wave = threadIdx.x >> 5;
    const int r = blockIdx.x * 8 + wave;

    float* Am = ws + OFF_A;
    float* Bw = ws + OFF_BW;
    float* eT = ws + OFF_ET;
    float* qT = ws + OFF_QT;
    float* aT = ws + OFF_AT;

    float cpart = 0.f, lqcpart = 0.f, lpzpart = 0.f;
    #pragma unroll
    for (int t = 0; t < 2; ++t) {
        const int d = lane + 32 * t;
        const float xv = x[r * DD + d];
        const float mv = mu[r * DD + d];
        const float lg = lv[r * DD + d];
        const float p = __expf(-lg);
        const float q = -0.5f * p;
        const float a = mv * p;
        const float e = -0.5f * (kLog2Pi + lg) - 0.5f * mv * mv * p;

        Am[r * 128 + d]      = xv * xv;
        Am[r * 128 + 64 + d] = xv;
        Bw[r * 128 + d]      = q;
        Bw[r * 128 + 64 + d] = a;
        eT[d * BB + r] = e;
        qT[d * BB + r] = q;
        aT[d * BB + r] = a;

        cpart += e;
        const float df = xv - mv;
        lqcpart += -0.5f * (kLog2Pi + lg) - 0.5f * df * df * p;
        lpzpart += -0.5f * kLog2Pi - 0.5f * xv * xv;
    }
    #pragma unroll
    for (int off = 16; off; off >>= 1) {
        cpart   += __shfl_xor(cpart, off, 32);
        lqcpart += __shfl_xor(lqcpart, off, 32);
        lpzpart += __shfl_xor(lpzpart, off, 32);
    }
    if (lane == 0) {
        ws[OFF_C + r]   = cpart;
        ws[OFF_LQC + r] = lqcpart;
        ws[OFF_LPZ + r] = lpzpart;
    }
}

// ---------------------------------------------------------------------------
// Kernel 2: S = [x^2|x] * [q|a]^T + c[j] + log_iw, via V_WMMA_F32_16X16X4_F32.
// One wave per 16x16 output tile, 8 tiles per 256-thread block.
// ---------------------------------------------------------------------------
__global__ void k_gemm(float* __restrict__ ws) {
    const int lane = threadIdx.x & 31;
    const int wave = threadIdx.x >> 5;
    const int tile = blockIdx.x * 8 + wave;
    const int ti = tile >> 7;    // tile / 128
    const int tj = tile & 127;

    const float* Am = ws + OFF_A;
    const float* Bw = ws + OFF_BW;
    const float* cv = ws + OFF_C;
    float* S = ws + OFF_S;

    const int rn = lane & 15;     // A: row m ; B: column n (same lane slot)
    const int kh = lane >> 4;     // K half: lanes 0-15 -> K{0,1}, 16-31 -> K{2,3}
    const float* ap = Am + (size_t)(ti * 16 + rn) * 128 + kh * 2;
    const float* bp = Bw + (size_t)(tj * 16 + rn) * 128 + kh * 2;

    v8f acc = {};
    #pragma unroll
    for (int kb = 0; kb < 128; kb += 4) {
        v2f a = *(const v2f*)(ap + kb);
        v2f b = *(const v2f*)(bp + kb);
        acc = __builtin_amdgcn_wmma_f32_16x16x4_f32(
            /*neg_a=*/false, a, /*neg_b=*/false, b,
            /*c_mod=*/(short)0, acc, /*reuse_a=*/false, /*reuse_b=*/false);
    }

    const float logInvN  = ws[OFF_LW + 0];
    const float logStrat = ws[OFF_LW + 1];

    const int j = tj * 16 + rn;
    const float cj = cv[j];
    #pragma unroll
    for (int v = 0; v < 8; ++v) {
        const int i = ti * 16 + v + 8 * kh;
        const float val = acc[v] + cj + ((i == j) ? logInvN : logStrat);
        S[(size_t)i * BB + j] = val;
    }
}

// ---------------------------------------------------------------------------
// Kernel 3: log_qz[i] = logsumexp_j S[i,j].  One 256-thread block per row.
// ---------------------------------------------------------------------------
__global__ void k_logqz(float* __restrict__ ws) {
    __shared__ float sm[256];
    __shared__ float ss[256];
    const int i = blockIdx.x;
    const int tid = threadIdx.x;
    const float* Srow = ws + OFF_S + (size_t)i * BB;

    float m = -INFINITY, s = 0.f;
    for (int j = tid; j < BB; j += 256) lse_push(m, s, Srow[j]);
    sm[tid] = m; ss[tid] = s;
    __syncthreads();
    for (int off = 128; off; off >>= 1) {
        if (tid < off) {
            float m1 = sm[tid], s1 = ss[tid];
            lse_merge(m1, s1, sm[tid + off], ss[tid + off]);
            sm[tid] = m1; ss[tid] = s1;
        }
        __syncthreads();
    }
    if (tid == 0) ws[OFF_LQZ + i] = sm[0] + logf(ss[0]);
}

// ---------------------------------------------------------------------------
// Kernel 4: log_prod_qzi[i] = sum_d logsumexp_j(e[j,d]+q[j,d]x2+a[j,d]x+log_iw).
// One block per i; each of 8 waves handles 8 dims with lane-strided online LSE.
// ---------------------------------------------------------------------------
__global__ void k_logprod(float* __restrict__ ws) {
    const int lane = threadIdx.x & 31;
    const int wave = threadIdx.x >> 5;
    const int i = blockIdx.x;

    const float logInvN  = ws[OFF_LW + 0];
    const float logStrat = ws[OFF_LW + 1];

    const float* Am = ws + OFF_A;
    __shared__ float partial[8];

    float acc = 0.f;
    for (int dd = 0; dd < 8; ++dd) {
        const int d = wave * 8 + dd;
        const float x2 = Am[(size_t)i * 128 + d];
        const float xv = Am[(size_t)i * 128 + 64 + d];
        const float* e = ws + OFF_ET + (size_t)d * BB;
        const float* q = ws + OFF_QT + (size_t)d * BB;
        const float* a = ws + OFF_AT + (size_t)d * BB;

        float m = -INFINITY, s = 0.f;
        for (int j = lane; j < BB; j += 32) {
            float v = fmaf(q[j], x2, fmaf(a[j], xv, e[j]));
            v += (j == i) ? logInvN : logStrat;
            lse_push(m, s, v);
        }
        #pragma unroll
        for (int off = 16; off; off >>= 1) {
            const float m2 = __shfl_xor(m, off, 32);
            const float s2 = __shfl_xor(s, off, 32);
            lse_merge(m, s, m2, s2);
        }
        acc += m + logf(s);
    }
    if (lane == 0) partial[wave] = acc;
    __syncthreads();
    if (threadIdx.x == 0) {
        float t = 0.f;
        #pragma unroll
        for (int w = 0; w < 8; ++w) t += partial[w];
        ws[OFF_LPQ + i] = t;
    }
}

// ---------------------------------------------------------------------------
// Kernel 5: final means -> out[0..2] = {mi_loss, tc_loss, dw_kl_loss}
// ---------------------------------------------------------------------------
__global__ void k_final(const float* __restrict__ ws, float* __restrict__ out) {
    __shared__ float smi[256], stc[256], sdw[256];
    const int tid = threadIdx.x;
    float mi = 0.f, tc = 0.f, dw = 0.f;
    for (int r = tid; r < BB; r += 256) {
        const float lqc = ws[OFF_LQC + r];
        const float lpz = ws[OFF_LPZ + r];
        const float lqz = ws[OFF_LQZ + r];
        const float lpq = ws[OFF_LPQ + r];
        mi += lqc - lqz;
        tc += lqz - lpq;
        dw += lpq - lpz;
    }
    smi[tid] = mi; stc[tid] = tc; sdw[tid] = dw;
    __syncthreads();
    for (int off = 128; off; off >>= 1) {
        if (tid < off) {
            smi[tid] += smi[tid + off];
            stc[tid] += stc[tid + off];
            sdw[tid] += sdw[tid + off];
        }
        __syncthreads();
    }
    if (tid == 0) {
        out[0] = smi[0] / (float)BB;
        out[1] = stc[0] / (float)BB;
        out[2] = sdw[0] / (float)BB;
    }
}

extern "C" void kernel_launch(void* const* d_in, const int* in_sizes, int n_in,
                              void* d_out, int out_size, void* d_ws, size_t ws_size,
                              hipStream_t stream) {
    const float* x  = (const float*)d_in[0];   // latent_sample (2048,64)
    const float* mu = (const float*)d_in[1];   // mu            (2048,64)
    const float* lv = (const float*)d_in[2];   // logvar        (2048,64)
    const int* dsz  = (const int*)d_in[3];     // data_size scalar
    float* out = (float*)d_out;
    float* ws  = (float*)d_ws;

    (void)in_sizes; (void)n_in; (void)out_size; (void)ws_size;

    // 1) per-row preprocessing (+ log-weight scalars): 2048 rows, 1 wave/row
    k_prep<<<BB / 8, 256, 0, stream>>>(x, mu, lv, dsz, ws);
    // 2) WMMA GEMM: 128x128 tiles, 8 tiles/block
    k_gemm<<<(128 * 128) / 8, 256, 0, stream>>>(ws);
    // 3) row logsumexp -> log_qz
    k_logqz<<<BB, 256, 0, stream>>>(ws);
    // 4) per-dim logsumexp -> log_prod_qzi
    k_logprod<<<BB, 256, 0, stream>>>(ws);
    // 5) final reduction -> 3 scalars
    k_final<<<1, 256, 0, stream>>>(ws, out);
}